// MeanVoxelEncoder_Radar_withDop_22119081575175
// MI455X (gfx1250) — compile-verified
//
#include <hip/hip_runtime.h>
#include <stdint.h>

// ---------------------------------------------------------------------------
// MeanVoxelEncoder_Radar: the reference collapses to "per sample, take the
// top-10% cube values (exact top-k semantics incl. index tie-break), emit the
// first 16000 of them in row-major order".  Memory-bound: 4 streaming reads of
// the 105MB cube ~= 18us at 23.3 TB/s.  CDNA5 paths used:
//   * global_load_async_to_lds_b128 + s_wait_asynccnt double-buffered staging
//     in the histogram passes (inline asm; portable across toolchains)
//   * v_wmma_f32_16x16x4_f32 exact 16-wide prefix scan in the block-scan stage
// ---------------------------------------------------------------------------

#define B_      8
#define ZDIM    32
#define YDIM    320
#define XDIM    320
#define NSAMP   (ZDIM * YDIM * XDIM)   // 3,276,800 elements / sample
#define KTOP    (NSAMP / 10)           // 327,680 points above 0.9 quantile
#define MAXVOX  16000
#define NBINS   65536
#define S_BLK   2048                   // elements per block in count/emit
#define NBLK    (NSAMP / S_BLK)        // 1600 blocks / sample
#define H_ITER  32
#define H_CHUNK 1024                   // floats staged per iteration (256 thr * 4)
#define H_PER_BLK (H_ITER * H_CHUNK)   // 32768
#define H_NBLK  (NSAMP / H_PER_BLK)    // 100 blocks / sample

// order-preserving float -> uint key (bigger float => bigger key)
__device__ __forceinline__ uint32_t fkey(float f) {
  uint32_t u = __float_as_uint(f);
  return u ^ ((uint32_t)((int32_t)u >> 31) | 0x80000000u);
}

// ---------------------------------------------------------------------------
// scratch hist zero (d_ws is poisoned, must clear before each histogram pass)
// ---------------------------------------------------------------------------
__global__ void zero_hist_kernel(uint32_t* __restrict__ hist) {
  uint32_t i = blockIdx.x * 256u + threadIdx.x;     // grid covers B_*NBINS
  hist[i] = 0u;
}

// ---------------------------------------------------------------------------
// Histogram pass (mode 0: high 16 bits of key; mode 1: low 16 bits restricted
// to the selected high bucket).  Input is staged through LDS with
// double-buffered async-to-LDS loads so the next chunk's HBM latency overlaps
// the LDS-read + atomic work of the current chunk.
// ---------------------------------------------------------------------------
__global__ void hist_pass_kernel(const float* __restrict__ src,
                                 uint32_t* __restrict__ hist,
                                 const uint32_t* __restrict__ state,
                                 int mode) {
  __shared__ float stage[2][H_CHUNK];
  const int tid = threadIdx.x;
  const int s   = blockIdx.y;
  const uint32_t selHigh = (mode == 1) ? state[s * 8 + 0] : 0u;
  uint32_t* h = hist + (size_t)s * NBINS;
  const float* base = src + (size_t)s * NSAMP + (size_t)blockIdx.x * H_PER_BLK;
  // low 32 bits of the generic LDS pointer == wave-relative LDS byte address
  const uint32_t ldsBase = (uint32_t)(uintptr_t)(&stage[0][0]);

  // issue one async 16B-per-lane load of iteration `it` into buffer `buf`
  auto issue = [&](int it, int buf) {
    uint64_t ga  = (uint64_t)(uintptr_t)(base + (size_t)it * H_CHUNK + tid * 4);
    uint32_t lds = ldsBase + (uint32_t)buf * (H_CHUNK * 4u) + (uint32_t)tid * 16u;
    asm volatile("global_load_async_to_lds_b128 %0, %1, off"
                 :: "v"(lds), "v"(ga) : "memory");
  };

  issue(0, 0);
  for (int it = 0; it < H_ITER; ++it) {
    const int cur = it & 1;
    if (it + 1 < H_ITER) {
      issue(it + 1, (it + 1) & 1);
      asm volatile("s_wait_asynccnt 0x1" ::: "memory");  // oldest load done
    } else {
      asm volatile("s_wait_asynccnt 0x0" ::: "memory");
    }
    __syncthreads();
    #pragma unroll
    for (int j = 0; j < 4; ++j) {
      float    v   = stage[cur][tid * 4 + j];
      uint32_t key = fkey(v);
      if (mode == 0) {
        atomicAdd(&h[key >> 16], 1u);
      } else if ((key >> 16) == selHigh) {
        atomicAdd(&h[key & 0xFFFFu], 1u);
      }
    }
    __syncthreads();   // buffer `cur` may be reused by issue(it+2)
  }
}

// ---------------------------------------------------------------------------
// Radix-select over 65536 bins: find bin b with cumAbove(b) < Kneed <=
// cumAbove(b)+hist[b], scanning from the largest key downward.
// round 0 -> state[0]=selHigh, state[1]=Kneed remaining inside bucket
// round 1 -> state[2]=T (exact K-th largest key), state[3]=q (tie quota)
// ---------------------------------------------------------------------------
__global__ void radix_select_kernel(const uint32_t* __restrict__ hist,
                                    uint32_t* __restrict__ state,
                                    int round) {
  __shared__ uint32_t cs[256];
  const int t = threadIdx.x;
  const int s = blockIdx.x;
  const uint32_t* h = hist + (size_t)s * NBINS;
  const uint32_t kneed = (round == 0) ? (uint32_t)KTOP : state[s * 8 + 1];

  uint32_t sum = 0;
  for (int j = 0; j < 256; ++j) sum += h[t * 256 + j];
  cs[t] = sum;
  __syncthreads();
  for (int off = 1; off < 256; off <<= 1) {           // inclusive scan
    uint32_t a = (t >= off) ? cs[t - off] : 0u;
    __syncthreads();
    cs[t] += a;
    __syncthreads();
  }
  const uint32_t total = cs[255];
  const uint32_t F = total - cs[t];                   // keys in chunks above t
  if (F < kneed && F + sum >= kneed) {                // unique owner thread
    uint32_t run = F, cumAbove = 0;
    int selected = 0;
    for (int j = 255; j >= 0; --j) {
      uint32_t hb = h[t * 256 + j];
      if (run + hb >= kneed) { selected = t * 256 + j; cumAbove = run; break; }
      run += hb;
    }
    if (round == 0) {
      state[s * 8 + 0] = (uint32_t)selected;
      state[s * 8 + 1] = kneed - cumAbove;
    } else {
      state[s * 8 + 2] = (state[s * 8 + 0] << 16) | (uint32_t)selected; // T
      state[s * 8 + 3] = kneed - cumAbove;                              // q
    }
  }
}

// ---------------------------------------------------------------------------
// Per-block (2048 elem) counts of key>T (G) and key==T (E)
// ---------------------------------------------------------------------------
__global__ void count_pass_kernel(const float* __restrict__ src,
                                  const uint32_t* __restrict__ state,
                                  uint32_t* __restrict__ bc) {
  __shared__ uint32_t sg[256], se[256];
  const int t = threadIdx.x, s = blockIdx.y, blk = blockIdx.x;
  const uint32_t T = state[s * 8 + 2];
  const float4* p = (const float4*)(src + (size_t)s * NSAMP + (size_t)blk * S_BLK) + t * 2;
  float4 a = p[0], b = p[1];
  float v[8] = {a.x, a.y, a.z, a.w, b.x, b.y, b.z, b.w};
  uint32_t g = 0, e = 0;
  #pragma unroll
  for (int j = 0; j < 8; ++j) { uint32_t k = fkey(v[j]); g += (k > T); e += (k == T); }
  sg[t] = g; se[t] = e;
  __syncthreads();
  for (int off = 128; off > 0; off >>= 1) {
    if (t < off) { sg[t] += sg[t + off]; se[t] += se[t + off]; }
    __syncthreads();
  }
  if (t == 0) {
    bc[((size_t)s * NBLK + blk) * 2 + 0] = sg[0];
    bc[((size_t)s * NBLK + blk) * 2 + 1] = se[0];
  }
}

// ---------------------------------------------------------------------------
// Exclusive scan of the 1600 (G,E) block-count pairs per sample.
// Middle stage (16 segment sums) is scanned with v_wmma_f32_16x16x4_f32:
//   P = L * V,  L = 16x16 lower-triangular ones, V[:,0]=G seg sums,
//   V[:,1]=E seg sums;  decomposed into 4 accumulating 16x16x4 WMMAs.
//   All values are integers < 2^24 -> exact in f32.
// ---------------------------------------------------------------------------
typedef float v2f __attribute__((ext_vector_type(2)));
typedef float v8f __attribute__((ext_vector_type(8)));

__global__ void scan_pass_kernel(const uint32_t* __restrict__ bc,
                                 uint32_t* __restrict__ bo) {
  __shared__ uint32_t sg[256], se[256];
  __shared__ float gg[16], ge[16];     // segment sums (16 groups of 16 threads)
  __shared__ float pg[16], pe[16];     // inclusive prefixes of segment sums
  const int t = threadIdx.x, s = blockIdx.x;
  const uint32_t* c = bc + (size_t)s * NBLK * 2;
  uint32_t* o = bo + (size_t)s * NBLK * 2;
  const int PT = 7;                    // 256*7 = 1792 >= 1600
  const int i0 = t * PT;

  uint32_t g = 0, e = 0;
  for (int j = 0; j < PT; ++j) {
    int i = i0 + j;
    if (i < NBLK) { g += c[i * 2]; e += c[i * 2 + 1]; }
  }
  sg[t] = g; se[t] = e;
  __syncthreads();
  if (t < 16) {
    uint32_t G = 0, E = 0;
    for (int j = 0; j < 16; ++j) { G += sg[t * 16 + j]; E += se[t * 16 + j]; }
    gg[t] = (float)G; ge[t] = (float)E;
  }
  __syncthreads();

#if __has_builtin(__builtin_amdgcn_wmma_f32_16x16x4_f32)
  if (t < 32) {                        // wave 0, full EXEC (WMMA requirement)
    const int lane = t, m = lane & 15, kb = (lane >> 4) * 2;
    v8f d = {0.f, 0.f, 0.f, 0.f, 0.f, 0.f, 0.f, 0.f};
    #pragma unroll
    for (int cchunk = 0; cchunk < 4; ++cchunk) {
      const int k0 = cchunk * 4 + kb, k1 = k0 + 1;
      v2f A, Bv;
      A.x = (k0 <= m) ? 1.0f : 0.0f;   // lower-triangular ones, columns k0,k1
      A.y = (k1 <= m) ? 1.0f : 0.0f;
      const int n = m;                 // B column index for this lane
      Bv.x = (n == 0) ? gg[k0] : ((n == 1) ? ge[k0] : 0.0f);
      Bv.y = (n == 0) ? gg[k1] : ((n == 1) ? ge[k1] : 0.0f);
      d = __builtin_amdgcn_wmma_f32_16x16x4_f32(false, A, false, Bv,
                                                (short)0, d, false, false);
    }
    // D[m][0] = G-prefix, D[m][1] = E-prefix  (C/D layout: lane=n + 16*(m>=8))
    if (lane == 0)  { for (int r = 0; r < 8; ++r) pg[r]     = d[r]; }
    if (lane == 16) { for (int r = 0; r < 8; ++r) pg[r + 8] = d[r]; }
    if (lane == 1)  { for (int r = 0; r < 8; ++r) pe[r]     = d[r]; }
    if (lane == 17) { for (int r = 0; r < 8; ++r) pe[r + 8] = d[r]; }
  }
#else
  if (t == 0) {
    float G = 0.f, E = 0.f;
    for (int j = 0; j < 16; ++j) { G += gg[j]; pg[j] = G; E += ge[j]; pe[j] = E; }
  }
#endif
  __syncthreads();

  const int grp = t >> 4;
  uint32_t exG = (grp == 0) ? 0u : (uint32_t)(pg[grp - 1] + 0.5f);
  uint32_t exE = (grp == 0) ? 0u : (uint32_t)(pe[grp - 1] + 0.5f);
  for (int j = grp * 16; j < t; ++j) { exG += sg[j]; exE += se[j]; }
  for (int j = 0; j < PT; ++j) {
    int i = i0 + j;
    if (i < NBLK) {
      o[i * 2 + 0] = exG;
      o[i * 2 + 1] = exE;
      exG += c[i * 2];
      exE += c[i * 2 + 1];
    }
  }
}

// ---------------------------------------------------------------------------
// Emit pass: exact ordered ranks; only ranks < MAXVOX write output rows.
// rank(elem) = #greater-before + min(#equal-before, q)
// ---------------------------------------------------------------------------
__device__ __forceinline__ void write_row(float* __restrict__ out, int s,
                                          uint32_t rank, int idx, float val) {
  const int z   = idx / (YDIM * XDIM);
  const int rem = idx % (YDIM * XDIM);
  const int y   = rem / XDIM;
  const int x   = rem % XDIM;
  const size_t row = (size_t)s * MAXVOX + rank;
  float* f = out + row * 4;                     // encoded features [B*16000,4]
  f[0] = x * 0.4f;
  f[1] = y * 0.4f - 64.0f;
  f[2] = z * 0.4f - 6.4f;
  f[3] = val * 1e-13f;
  float* cc = out + (size_t)B_ * MAXVOX * 4 + row * 4;  // coords (b,z,y,x)
  cc[0] = (float)s; cc[1] = (float)z; cc[2] = (float)y; cc[3] = (float)x;
  out[(size_t)B_ * MAXVOX * 8 + row] = 1.0f;    // num_points (always 1 here)
}

__global__ void emit_pass_kernel(const float* __restrict__ src,
                                 const uint32_t* __restrict__ state,
                                 const uint32_t* __restrict__ bo,
                                 float* __restrict__ out) {
  __shared__ uint32_t sg[256], se[256];
  const int t = threadIdx.x, s = blockIdx.y, blk = blockIdx.x;
  const uint32_t T = state[s * 8 + 2];
  const uint32_t q = state[s * 8 + 3];
  const uint32_t baseG = bo[((size_t)s * NBLK + blk) * 2 + 0];
  const uint32_t baseE = bo[((size_t)s * NBLK + blk) * 2 + 1];
  const uint32_t baseRank = baseG + (baseE < q ? baseE : q);
  if (baseRank >= MAXVOX) return;               // whole block past row 16000

  const float4* p = (const float4*)(src + (size_t)s * NSAMP + (size_t)blk * S_BLK) + t * 2;
  float4 a = p[0], b = p[1];
  float v[8] = {a.x, a.y, a.z, a.w, b.x, b.y, b.z, b.w};
  uint32_t k[8];
  uint32_t g = 0, e = 0;
  #pragma unroll
  for (int j = 0; j < 8; ++j) { k[j] = fkey(v[j]); g += (k[j] > T); e += (k[j] == T); }
  sg[t] = g; se[t] = e;
  __syncthreads();
  for (int off = 1; off < 256; off <<= 1) {     // inclusive scan of pairs
    uint32_t ag = 0, ae = 0;
    if (t >= off) { ag = sg[t - off]; ae = se[t - off]; }
    __syncthreads();
    sg[t] += ag; se[t] += ae;
    __syncthreads();
  }
  uint32_t runG = baseG + (sg[t] - g);          // exclusive for this thread
  uint32_t runE = baseE + (se[t] - e);

  #pragma unroll
  for (int j = 0; j < 8; ++j) {
    const int idx = blk * S_BLK + t * 8 + j;
    if (k[j] > T) {
      uint32_t rank = runG + (runE < q ? runE : q);
      if (rank < MAXVOX) write_row(out, s, rank, idx, v[j]);
      runG++;
    } else if (k[j] == T) {
      if (runE < q) {                           // tie quota -> smallest indices
        uint32_t rank = runG + runE;
        if (rank < MAXVOX) write_row(out, s, rank, idx, v[j]);
      }
      runE++;
    }
  }
}

// ---------------------------------------------------------------------------
extern "C" void kernel_launch(void* const* d_in, const int* in_sizes, int n_in,
                              void* d_out, int out_size, void* d_ws, size_t ws_size,
                              hipStream_t stream) {
  (void)in_sizes; (void)n_in; (void)out_size; (void)ws_size;
  const float* src = (const float*)d_in[0];
  float* out = (float*)d_out;

  uint8_t* ws = (uint8_t*)d_ws;
  uint32_t* hist  = (uint32_t*)(ws);                                   // 8*65536*4 = 2MB
  uint32_t* state = (uint32_t*)(ws + (size_t)B_ * NBINS * 4);          // 8*8 u32
  uint32_t* bc    = (uint32_t*)(ws + (size_t)B_ * NBINS * 4 + 1024);   // 8*1600*2 u32
  uint32_t* bo    = (uint32_t*)(ws + (size_t)B_ * NBINS * 4 + 1024 +
                                (size_t)B_ * NBLK * 2 * 4);

  const dim3 hgrid(H_NBLK, B_);
  const dim3 sgrid(NBLK, B_);

  zero_hist_kernel<<<(B_ * NBINS) / 256, 256, 0, stream>>>(hist);
  hist_pass_kernel<<<hgrid, 256, 0, stream>>>(src, hist, state, 0);
  radix_select_kernel<<<B_, 256, 0, stream>>>(hist, state, 0);
  zero_hist_kernel<<<(B_ * NBINS) / 256, 256, 0, stream>>>(hist);
  hist_pass_kernel<<<hgrid, 256, 0, stream>>>(src, hist, state, 1);
  radix_select_kernel<<<B_, 256, 0, stream>>>(hist, state, 1);
  count_pass_kernel<<<sgrid, 256, 0, stream>>>(src, state, bc);
  scan_pass_kernel<<<B_, 256, 0, stream>>>(bc, bo);
  emit_pass_kernel<<<sgrid, 256, 0, stream>>>(src, state, bo, out);
}